// GraphEncoder_6528350290582
// MI455X (gfx1250) — compile-verified
//
#include <hip/hip_runtime.h>
#include <hip/hip_bf16.h>
#include <math.h>

typedef __attribute__((ext_vector_type(2))) float v2f;
typedef __attribute__((ext_vector_type(8))) float v8f;

#define D_DIM 256
#define MT 5        // M-tiles per wave (V/16 = 3125 = 5 * 625)

// ---------------- Phase 0: zero the segment-sum accumulator ----------------
__global__ void zero_f4_kernel(float4* __restrict__ p, int n4) {
    int i = blockIdx.x * blockDim.x + threadIdx.x;
    if (i < n4) p[i] = make_float4(0.f, 0.f, 0.f, 0.f);
}

// ---------------- Phase 1: gather + weighted scatter-add (segment_sum) -----
// 64 threads per edge; each thread handles one float4 chunk of the 256-wide row.
// vrepr (51 MB) and acc (51 MB) both fit in the 192 MB L2, so gathers and
// the global_atomic_add_f32 scatter run at L2 speed.
__global__ void edge_scatter_kernel(const float* __restrict__ vrepr,
                                    const int*   __restrict__ sidx,
                                    const int*   __restrict__ tidx,
                                    const float* __restrict__ esgn,
                                    const float* __restrict__ enorm,
                                    float*       __restrict__ acc,
                                    int ne) {
    int t = blockIdx.x * blockDim.x + threadIdx.x;
    int e = t >> 6;                 // D/4 = 64 chunks per edge
    if (e >= ne) return;
    int chunk = (t & 63) << 2;      // float offset within row
    int s = sidx[e];
    int g = tidx[e];
    float w = enorm[e] * esgn[e];
    const float4 v = *reinterpret_cast<const float4*>(vrepr + (size_t)s * D_DIM + chunk);
    float* dst = acc + (size_t)g * D_DIM + chunk;
    atomicAdd(dst + 0, v.x * w);
    atomicAdd(dst + 1, v.y * w);
    atomicAdd(dst + 2, v.z * w);
    atomicAdd(dst + 3, v.w * w);
}

// ---------------- Phase 2: fused dual GEMM + softplus + rsample ------------
// One wave32 computes MT=5 vertically-stacked 16x16 output tiles for BOTH
// heads (one N-tile) with V_WMMA_F32_16X16X4_F32. Per K-step: 2 B loads are
// reused across 5 A tiles -> 7 float2 loads feed 10 WMMAs, and the 10
// independent accumulator chains pipeline well.
//
// A (16x4 f32, ISA layout): lane l<16 -> VGPR0=A[l][k], VGPR1=A[l][k+1];
//                           lane 16+l -> A[l][k+2], A[l][k+3]
// B (4x16): B[k][n] = W[n][k]  (y = x @ W.T), mirrored layout.
__global__ void __launch_bounds__(256)
gemm_heads_kernel(const float* __restrict__ acc,
                  const float* __restrict__ loc_w,
                  const float* __restrict__ loc_b,
                  const float* __restrict__ std_w,
                  const float* __restrict__ std_b,
                  const float* __restrict__ eps,
                  float*       __restrict__ out,   // [3, V, D] flat
                  int V, int ntiles) {
    const int lane = threadIdx.x & 31;
    const int wave = threadIdx.x >> 5;
    const int tile = blockIdx.x * 8 + wave;      // one tile = 80 rows x 16 cols
    if (tile >= ntiles) return;                  // wave-uniform guard
    const int ntn  = D_DIM / 16;                 // 16 N-tiles
    const int m0   = (tile / ntn) * (16 * MT);   // 80-row group
    const int n0   = (tile % ntn) * 16;

    const int half = lane >> 4;                  // 0: K=0,1 ; 1: K=2,3
    const int lidx = lane & 15;

    const float* Ap[MT];
#pragma unroll
    for (int i = 0; i < MT; ++i)
        Ap[i] = acc + (size_t)(m0 + i * 16 + lidx) * D_DIM + 2 * half;
    const float* BLp = loc_w + (size_t)(n0 + lidx) * D_DIM + 2 * half;
    const float* BSp = std_w + (size_t)(n0 + lidx) * D_DIM + 2 * half;

    v8f cl[MT];
    v8f cs[MT];
#pragma unroll
    for (int i = 0; i < MT; ++i) { cl[i] = (v8f){}; cs[i] = (v8f){}; }

#pragma unroll 2
    for (int k = 0; k < D_DIM; k += 4) {
        v2f bl = *(const v2f*)(BLp + k);
        v2f bs = *(const v2f*)(BSp + k);
        v2f a[MT];
#pragma unroll
        for (int i = 0; i < MT; ++i) a[i] = *(const v2f*)(Ap[i] + k);
#pragma unroll
        for (int i = 0; i < MT; ++i) {
            // 8 args: (neg_a, A, neg_b, B, c_mod, C, reuse_a, reuse_b)
            cl[i] = __builtin_amdgcn_wmma_f32_16x16x4_f32(false, a[i], false, bl,
                                                          (short)0, cl[i], false, false);
            cs[i] = __builtin_amdgcn_wmma_f32_16x16x4_f32(false, a[i], false, bs,
                                                          (short)0, cs[i], false, false);
        }
    }

    // Epilogue per ISA C/D layout: VGPR v -> row base+v (lanes 0-15) or base+v+8
    const int col = n0 + lidx;
    const float lb = loc_b[col];
    const float sb = std_b[col];
    const size_t VD = (size_t)V * D_DIM;
#pragma unroll
    for (int i = 0; i < MT; ++i) {
#pragma unroll
        for (int v = 0; v < 8; ++v) {
            const int row = m0 + i * 16 + v + half * 8;
            const size_t o = (size_t)row * D_DIM + col;
            float loc = cl[i][v] + lb;
            float x   = cs[i][v] + sb;
            float sp  = (x > 20.f) ? x : log1pf(expf(x));   // softplus
            sp += 1e-7f;
            float vs = loc + sp * eps[o];
            out[o]          = loc;
            out[VD + o]     = sp;
            out[2 * VD + o] = vs;
        }
    }
}

extern "C" void kernel_launch(void* const* d_in, const int* in_sizes, int n_in,
                              void* d_out, int out_size, void* d_ws, size_t ws_size,
                              hipStream_t stream) {
    const float* vrepr = (const float*)d_in[0];
    const int*   sidx  = (const int*)  d_in[1];
    const int*   tidx  = (const int*)  d_in[2];
    const float* esgn  = (const float*)d_in[3];
    const float* enorm = (const float*)d_in[4];
    const float* loc_w = (const float*)d_in[5];
    const float* loc_b = (const float*)d_in[6];
    const float* std_w = (const float*)d_in[7];
    const float* std_b = (const float*)d_in[8];
    const float* eps   = (const float*)d_in[9];
    float* out = (float*)d_out;
    float* acc = (float*)d_ws;                 // [V, D] f32 accumulator

    const int Dd = in_sizes[6];                // 256 (loc_b length)
    const int V  = in_sizes[0] / Dd;           // 50000
    const int E  = in_sizes[1];                // 1,600,000

    // Phase 0: zero accumulator (every call; harness replays the graph)
    int n4 = (V * Dd) / 4;
    zero_f4_kernel<<<(n4 + 255) / 256, 256, 0, stream>>>((float4*)acc, n4);

    // Phase 1: edge gather/scatter
    long long sthreads = (long long)E * (Dd / 4);
    int sblocks = (int)((sthreads + 255) / 256);
    edge_scatter_kernel<<<sblocks, 256, 0, stream>>>(vrepr, sidx, tidx, esgn,
                                                     enorm, acc, E);

    // Phase 2: dual-head WMMA GEMM + epilogue.
    // tiles = (V / (16*MT)) * (Dd/16) = 625 * 16 = 10000; 8 waves per block.
    int ntiles = (V / (16 * MT)) * (Dd / 16);
    gemm_heads_kernel<<<(ntiles + 7) / 8, 256, 0, stream>>>(acc, loc_w, loc_b,
                                                            std_w, std_b, eps,
                                                            out, V, ntiles);
}